// ShiftingModule_79388175499469
// MI455X (gfx1250) — compile-verified
//
#include <hip/hip_runtime.h>
#include <hip/hip_bf16.h>
#include <math.h>

typedef __attribute__((ext_vector_type(2))) float v2f;
typedef __attribute__((ext_vector_type(8))) float v8f;

#define B_DIM  4096
#define S_DIM  1728
#define C_DIM  24
#define PREDL  24
#define NWIN   72      // S/PRED_LEN
#define IN_DIM 3504    // (2 + 2*NWIN) * C
#define HID    512
#define EPSV   1e-5f

// -------------------------------------------------------------------------
// Kernel 1: windowed mean/std over the time-flipped series -> feat rows 0..143
// One wave per flipped window i; lanes 0..23 = channels. Flipped window i of
// flip(x) corresponds to original rows [S - 24*(i+1), S - 24*i).
// -------------------------------------------------------------------------
__global__ __launch_bounds__(256) void stats_kernel(const float* __restrict__ x,
                                                    float* __restrict__ feat) {
  const int b       = blockIdx.x / 9;          // 9 blocks * 8 waves = 72 windows
  const int winBase = (blockIdx.x % 9) * 8;
  const int wave    = threadIdx.x >> 5;
  const int lane    = threadIdx.x & 31;
  const int i       = winBase + wave;          // flipped window index 0..71
  if (lane >= C_DIM) return;

  const int sStart = S_DIM - PREDL * (i + 1);
  const float* xp = x + ((size_t)b * S_DIM + sStart) * C_DIM + lane;

  float sum = 0.0f, sumsq = 0.0f;
#pragma unroll
  for (int j = 0; j < PREDL; ++j) {
    float v = xp[j * C_DIM];
    sum   += v;
    sumsq += v * v;
  }
  const float mean = sum * (1.0f / PREDL);
  const float var  = sumsq * (1.0f / PREDL) - mean * mean;
  const float stdv = sqrtf(var + EPSV);

  float* fb = feat + (size_t)b * IN_DIM;
  fb[i * C_DIM + lane]          = mean;  // rows 0..71  : hist_mean
  fb[(NWIN + i) * C_DIM + lane] = stdv;  // rows 72..143: hist_std
}

// -------------------------------------------------------------------------
// Kernel 2: append stats1 (row 144) and stats2 (row 145) to feat
// -------------------------------------------------------------------------
__global__ void tail_kernel(const float* __restrict__ s1,
                            const float* __restrict__ s2,
                            float* __restrict__ feat) {
  const int b = blockIdx.x;
  const int t = threadIdx.x;
  float* fb = feat + (size_t)b * IN_DIM + (2 * NWIN) * C_DIM;
  if (t < C_DIM)          fb[t] = s1[b * C_DIM + t];
  else if (t < 2 * C_DIM) fb[t] = s2[b * C_DIM + (t - C_DIM)];
}

// -------------------------------------------------------------------------
// Kernel 3: h[4096,512] = feat[4096,3504] @ W1[3504,512]
// fp32 WMMA 16x16x4. Block = 8 waves (2x4) each owning a 2x2 grid of 16x16
// tiles (wave footprint 32x32) -> block tile 64(M) x 128(N). One pair of
// A-frags + one pair of B-frags feeds 4 WMMAs, halving LDS traffic per WMMA.
// K staged through LDS in float4 chunks of 48 (3504 = 73*48).
// -------------------------------------------------------------------------
#define BM 64
#define BN 128
#define KC 48

__global__ __launch_bounds__(256) void gemm1_kernel(const float* __restrict__ feat,
                                                    const float* __restrict__ W1,
                                                    float* __restrict__ h) {
  __shared__ __align__(16) float As[BM][KC];   // 12 KB
  __shared__ __align__(16) float Bs[KC][BN];   // 24 KB

  const int tileM = blockIdx.x * BM;
  const int tileN = blockIdx.y * BN;
  const int wave  = threadIdx.x >> 5;
  const int lane  = threadIdx.x & 31;
  const int wmB   = (wave >> 2) * 32;   // 0 or 32   : wave M base in block tile
  const int wnB   = (wave & 3) * 32;    // 0..96     : wave N base in block tile
  const int m     = lane & 15;
  const int kh    = (lane >> 4) * 2;    // lanes 0-15 -> K0/K1, lanes 16-31 -> K2/K3

  v8f acc[2][2] = {};

  for (int k0 = 0; k0 < IN_DIM; k0 += KC) {
    // ---- cooperative float4 stage: A (64x48) = 3 vec4/thread -------------
#pragma unroll
    for (int r = 0; r < (BM * KC) / (256 * 4); ++r) {      // 3
      int idx  = threadIdx.x + r * 256;                    // vec4 index
      int row  = idx / (KC / 4);
      int colv = idx % (KC / 4);
      float4 v = *reinterpret_cast<const float4*>(
          &feat[(size_t)(tileM + row) * IN_DIM + k0 + colv * 4]);
      *reinterpret_cast<float4*>(&As[row][colv * 4]) = v;
    }
    // ---- cooperative float4 stage: B (48x128) = 6 vec4/thread ------------
#pragma unroll
    for (int r = 0; r < (KC * BN) / (256 * 4); ++r) {      // 6
      int idx  = threadIdx.x + r * 256;
      int row  = idx / (BN / 4);
      int colv = idx % (BN / 4);
      float4 v = *reinterpret_cast<const float4*>(
          &W1[(size_t)(k0 + row) * HID + tileN + colv * 4]);
      *reinterpret_cast<float4*>(&Bs[row][colv * 4]) = v;
    }
    __syncthreads();

#pragma unroll
    for (int kk = 0; kk < KC; kk += 4) {
      v2f a0, a1, b0, b1;
      // A frags (16x4): VGPR0=K(kh), VGPR1=K(kh+1) for this half-wave
      a0.x = As[wmB + m][kk + kh];
      a0.y = As[wmB + m][kk + kh + 1];
      a1.x = As[wmB + 16 + m][kk + kh];
      a1.y = As[wmB + 16 + m][kk + kh + 1];
      // B frags (4x16): same K split across half-waves, lane = column
      b0.x = Bs[kk + kh][wnB + m];
      b0.y = Bs[kk + kh + 1][wnB + m];
      b1.x = Bs[kk + kh][wnB + 16 + m];
      b1.y = Bs[kk + kh + 1][wnB + 16 + m];

      acc[0][0] = __builtin_amdgcn_wmma_f32_16x16x4_f32(false, a0, false, b0,
                                                        (short)0, acc[0][0], false, false);
      acc[0][1] = __builtin_amdgcn_wmma_f32_16x16x4_f32(false, a0, false, b1,
                                                        (short)0, acc[0][1], false, false);
      acc[1][0] = __builtin_amdgcn_wmma_f32_16x16x4_f32(false, a1, false, b0,
                                                        (short)0, acc[1][0], false, false);
      acc[1][1] = __builtin_amdgcn_wmma_f32_16x16x4_f32(false, a1, false, b1,
                                                        (short)0, acc[1][1], false, false);
    }
    __syncthreads();
  }

  // C/D layout: VGPR r -> row (r) for lanes 0-15, row (r+8) for lanes 16-31.
#pragma unroll
  for (int i = 0; i < 2; ++i) {
#pragma unroll
    for (int j = 0; j < 2; ++j) {
      const int rowBase = tileM + wmB + i * 16 + (lane >> 4) * 8;
      const int col     = tileN + wnB + j * 16 + m;
#pragma unroll
      for (int r = 0; r < 8; ++r) {
        h[(size_t)(rowBase + r) * HID + col] = acc[i][j][r];
      }
    }
  }
}

// -------------------------------------------------------------------------
// Kernel 4: y[4096,2] = h[4096,512] @ W2[512,2]
// -------------------------------------------------------------------------
__global__ __launch_bounds__(256) void gemm2_kernel(const float* __restrict__ h,
                                                    const float* __restrict__ W2,
                                                    float* __restrict__ y) {
  __shared__ float red0[256];
  __shared__ float red1[256];
  const int b = blockIdx.x;
  const int t = threadIdx.x;

  float a0 = 0.0f, a1 = 0.0f;
  const float* hb = h + (size_t)b * HID;
  for (int k = t; k < HID; k += 256) {
    float v = hb[k];
    a0 += v * W2[k * 2 + 0];
    a1 += v * W2[k * 2 + 1];
  }
  red0[t] = a0;
  red1[t] = a1;
  __syncthreads();
  for (int s = 128; s > 0; s >>= 1) {
    if (t < s) {
      red0[t] += red0[t + s];
      red1[t] += red1[t + s];
    }
    __syncthreads();
  }
  if (t == 0) {
    y[b * 2 + 0] = red0[0];
    y[b * 2 + 1] = red1[0];
  }
}

// -------------------------------------------------------------------------
extern "C" void kernel_launch(void* const* d_in, const int* in_sizes, int n_in,
                              void* d_out, int out_size, void* d_ws, size_t ws_size,
                              hipStream_t stream) {
  const float* x  = (const float*)d_in[0];   // [4096,1728,24]
  const float* s1 = (const float*)d_in[1];   // [4096,1,24]
  const float* s2 = (const float*)d_in[2];   // [4096,1,24]
  const float* W1 = (const float*)d_in[3];   // [3504,512]
  const float* W2 = (const float*)d_in[4];   // [512,2]

  float* feat = (float*)d_ws;                        // 4096*3504 floats
  float* hbuf = feat + (size_t)B_DIM * IN_DIM;       // 4096*512 floats
  float* y    = (float*)d_out;                       // [4096,2]

  stats_kernel<<<B_DIM * 9, 256, 0, stream>>>(x, feat);
  tail_kernel<<<B_DIM, 64, 0, stream>>>(s1, s2, feat);

  dim3 g1(B_DIM / BM, HID / BN);                     // 64 x 4
  gemm1_kernel<<<g1, 256, 0, stream>>>(feat, W1, hbuf);

  gemm2_kernel<<<B_DIM, 256, 0, stream>>>(hbuf, W2, y);
}